// PCF_20255065768438
// MI455X (gfx1250) — compile-verified
//
#include <hip/hip_runtime.h>
#include <hip/hip_bf16.h>

// PCF guided point-convolution for MI455X (gfx1250, wave32).
// Per point: D[64x16] = (gathered*guidance)^T [64x16] x weightnet [16x16]
// -> 16x V_WMMA_F32_16X16X4_F32 per point, one wave32 per point.
// Output-bandwidth bound (~480 MB total, ~21 us @ 23.3 TB/s); this version
// trims vmem issue: guidance rows via broadcast b128 loads + cndmask select,
// neighbor index pairs via b64 loads.

typedef __attribute__((ext_vector_type(2))) float v2f;
typedef __attribute__((ext_vector_type(4))) float v4f;
typedef __attribute__((ext_vector_type(8))) float v8f;
typedef __attribute__((ext_vector_type(2))) int   v2i;

#define PCF_B 2
#define PCF_N 40000
#define PCF_M 40000
#define PCF_K 16
#define PCF_C 64
#define PCF_H 8
#define PCF_D 16   // C_MID

__global__ __launch_bounds__(256) void pcf_wmma_kernel(
    const float* __restrict__ feats,  // [B, N, 64]
    const int*   __restrict__ inds,   // [B, M, 16]
    const float* __restrict__ guide,  // [B, M, 16, 8]
    const float* __restrict__ wnet,   // [B, M, 16, 16]
    float*       __restrict__ out)    // [B, M, 1024]
{
    const int lane = threadIdx.x & 31;
    const int wave = threadIdx.x >> 5;
    const long p = (long)blockIdx.x * 8 + wave;          // flat point id
    if (p >= (long)PCF_B * PCF_M) return;                // wave-uniform guard

    const int  b       = (int)(p / PCF_M);
    const float* fb    = feats + (long)b * PCF_N * PCF_C;
    const long pbase16 = p * PCF_K;                      // base for [.., k] dims

    const int half = lane >> 4;          // 0: lanes 0-15, 1: lanes 16-31
    const int l15  = lane & 15;
    const bool hsel = (l15 & 8) != 0;    // h0 = l15>>3: odd/even head select

    // ---- B-matrix chunks: weightnet[k0..k3, 0..15] per kc (reused for all cb)
    // 32-bit B 4x16 layout (mirror of A): lanes 0-15 = N cols, VGPR0 holds
    // K=0 (lanes 0-15) / K=2 (lanes 16-31); VGPR1 holds K=1 / K=3.
    v2f Bm[4];
#pragma unroll
    for (int kc = 0; kc < 4; ++kc) {
        const int k0 = kc * 4 + half * 2;
        const float* wp = wnet + (pbase16 + k0) * PCF_D + l15;
        Bm[kc].x = wp[0];
        Bm[kc].y = wp[PCF_D];   // k0+1 row
    }

    v8f acc[4] = {v8f{}, v8f{}, v8f{}, v8f{}};

#pragma unroll
    for (int kc = 0; kc < 4; ++kc) {
        const int k0 = kc * 4 + half * 2;     // this lane's K pair
        // neighbor indices k0,k0+1 are adjacent -> single b64 load
        const v2i ii = *(const v2i*)(inds + pbase16 + k0);
        const float* f0 = fb + (long)ii.x * PCF_C;
        const float* f1 = fb + (long)ii.y * PCF_C;

        // guidance rows for k0 and k0+1: 8 floats each, loaded as 2x b128.
        // Address is lane-uniform per half-wave -> broadcast loads.
        const float* gp0 = guide + (pbase16 + k0) * PCF_H;
        const v4f gl0 = *(const v4f*)(gp0);          // heads 0-3, k0
        const v4f gh0 = *(const v4f*)(gp0 + 4);      // heads 4-7, k0
        const v4f gl1 = *(const v4f*)(gp0 + 8);      // heads 0-3, k0+1
        const v4f gh1 = *(const v4f*)(gp0 + 12);     // heads 4-7, k0+1
        // head for (cb, lane) = 2*cb + h0 -> component select by h0
        float g0s[4], g1s[4];
        g0s[0] = hsel ? gl0.y : gl0.x;   // head 0+h0
        g0s[1] = hsel ? gl0.w : gl0.z;   // head 2+h0
        g0s[2] = hsel ? gh0.y : gh0.x;   // head 4+h0
        g0s[3] = hsel ? gh0.w : gh0.z;   // head 6+h0
        g1s[0] = hsel ? gl1.y : gl1.x;
        g1s[1] = hsel ? gl1.w : gl1.z;
        g1s[2] = hsel ? gh1.y : gh1.x;
        g1s[3] = hsel ? gh1.w : gh1.z;

#pragma unroll
        for (int cb = 0; cb < 4; ++cb) {
            const int c = cb * 16 + l15;          // channel this lane owns
            // A-matrix 16x4 f32: lane<16 -> {K=k0,k0+1}, lane>=16 -> {K=k0+2,+3}
            v2f A;
            A.x = f0[c] * g0s[cb];
            A.y = f1[c] * g1s[cb];
            acc[cb] = __builtin_amdgcn_wmma_f32_16x16x4_f32(
                /*neg_a=*/false, A, /*neg_b=*/false, Bm[kc],
                /*c_mod=*/(short)0, acc[cb],
                /*reuse_a=*/false, /*reuse_b=*/false);
        }
    }

    // ---- Store D tiles: VGPR r holds row r (lanes 0-15) / row 8+r (lanes 16-31)
    // out[p, c*16 + d], c = cb*16 + half*8 + r, d = l15. Streaming output ->
    // non-temporal stores (write-once, 328 MB dominates traffic).
    float* op = out + p * (PCF_C * PCF_D);
#pragma unroll
    for (int cb = 0; cb < 4; ++cb) {
#pragma unroll
        for (int r = 0; r < 8; ++r) {
            const int c = cb * 16 + half * 8 + r;
            __builtin_nontemporal_store(acc[cb][r], op + c * PCF_D + l15);
        }
    }
}

extern "C" void kernel_launch(void* const* d_in, const int* in_sizes, int n_in,
                              void* d_out, int out_size, void* d_ws, size_t ws_size,
                              hipStream_t stream) {
    const float* feats = (const float*)d_in[0];   // [B, N, 64]
    const int*   inds  = (const int*)  d_in[1];   // [B, M, 16]
    const float* guide = (const float*)d_in[2];   // [B, M, 16, 8]
    const float* wnet  = (const float*)d_in[3];   // [B, M, 16, 16]
    float*       out   = (float*)d_out;           // [B, M, 1024]

    const long n_points = (long)PCF_B * PCF_M;    // 80000 -> 1 wave32 each
    const int  block    = 256;                    // 8 waves/block
    const int  grid     = (int)((n_points + 7) / 8);

    pcf_wmma_kernel<<<grid, block, 0, stream>>>(feats, inds, guide, wnet, out);
}